// PairwiseRankingLoss_60825326846106
// MI455X (gfx1250) — compile-verified
//
#include <hip/hip_runtime.h>
#include <math.h>

typedef __attribute__((ext_vector_type(2))) float v2f;
typedef __attribute__((ext_vector_type(8))) float v8f;

#define NCLS    32
#define BTOT    32768
#define DDIM    128
#define MBLK    1024
#define NPAIR   31
#define NCHUNK  256
#define CHUNKSZ 128
#define TILE    128
#define LSTRIDE 132   // 128 + 4 pad: keeps float4 stores 16B aligned, de-conflicts banks

// ---------------------------------------------------------------------------
// Kernel 1: per-chunk class histogram (256 chunks x 128 elems, 1 block).
// Each thread owns its own 32-entry LDS row -> no atomics, fully deterministic.
// ---------------------------------------------------------------------------
__global__ __launch_bounds__(256) void k_hist(const int* __restrict__ labels,
                                              int* __restrict__ cc) {
  __shared__ int h[NCHUNK * NCLS];
  const int t = threadIdx.x;
  for (int c = 0; c < NCLS; ++c) h[t * NCLS + c] = 0;
  const int base = t * CHUNKSZ;
  for (int e = 0; e < CHUNKSZ; ++e) {
    int lab = labels[base + e] & (NCLS - 1);
    h[t * NCLS + lab]++;
  }
  for (int c = 0; c < NCLS; ++c) cc[t * NCLS + c] = h[t * NCLS + c];
}

// ---------------------------------------------------------------------------
// Kernel 2: scan chunk histograms -> per-chunk/per-class scatter bases.
// 32 threads (one per class); thread 0 does the 32-class exclusive scan.
// ---------------------------------------------------------------------------
__global__ void k_scan(const int* __restrict__ cc,
                       int* __restrict__ cb,
                       int* __restrict__ cls) {
  const int c = threadIdx.x;  // 0..31
  int run = 0;
  for (int ch = 0; ch < NCHUNK; ++ch) {
    cb[ch * NCLS + c] = run;
    run += cc[ch * NCLS + c];
  }
  cls[c] = run;  // class totals
  __syncthreads();
  if (c == 0) {
    int acc = 0;
    for (int i = 0; i < NCLS; ++i) { int v = cls[i]; cls[i] = acc; acc += v; }
  }
  __syncthreads();
  const int base = cls[c];
  for (int ch = 0; ch < NCHUNK; ++ch) cb[ch * NCLS + c] += base;
}

// ---------------------------------------------------------------------------
// Kernel 3: stable scatter. Thread t sequentially walks chunk t; it owns
// cb[t*32 .. t*32+31] exclusively -> stable counting sort, no races.
// ---------------------------------------------------------------------------
__global__ __launch_bounds__(256) void k_scatter(const int* __restrict__ labels,
                                                 int* __restrict__ cb,
                                                 int* __restrict__ perm) {
  const int t = threadIdx.x;
  const int base = t * CHUNKSZ;
  for (int e = 0; e < CHUNKSZ; ++e) {
    const int i = base + e;
    const int lab = labels[i] & (NCLS - 1);
    const int pos = cb[t * NCLS + lab];
    cb[t * NCLS + lab] = pos + 1;
    perm[pos] = i;
  }
}

// ---------------------------------------------------------------------------
// Kernel 4: main pair-distance loss via V_WMMA_F32_16X16X4_F32.
// grid = (8, 8, 31): z = block pair, (y,x) = 128x128 macro-tile of the
// 1024x1024 distance matrix. 8 waves per WG, each wave -> 16x128 strip.
// ---------------------------------------------------------------------------
__global__ __launch_bounds__(256) void k_pairloss(const float* __restrict__ reps,
                                                  const int* __restrict__ perm,
                                                  float* __restrict__ partials) {
  __shared__ float sA[TILE * LSTRIDE];
  __shared__ float sB[TILE * LSTRIDE];
  __shared__ float nA[TILE];
  __shared__ float nB[TILE];
  __shared__ float rbuf[256];

  const int t = threadIdx.x;
  const int p = blockIdx.z;                    // pair 0..30
  const int aBase = p * MBLK + blockIdx.y * TILE;        // sorted row of A strip
  const int bBase = (p + 1) * MBLK + blockIdx.x * TILE;  // sorted row of B strip

  // ---- stage 128x128 A and B tiles (gather rows through perm) ----
  {
    const int row = t >> 1;
    const int colBase = (t & 1) * 64;
    const float* ga = reps + (size_t)perm[aBase + row] * DDIM;
    const float* gb = reps + (size_t)perm[bBase + row] * DDIM;
    float* la = &sA[row * LSTRIDE + colBase];
    float* lb = &sB[row * LSTRIDE + colBase];
#pragma unroll
    for (int i = 0; i < 16; ++i) {
      float4 va = *(const float4*)(ga + colBase + i * 4);
      float4 vb = *(const float4*)(gb + colBase + i * 4);
      *(float4*)(la + i * 4) = va;
      *(float4*)(lb + i * 4) = vb;
    }
  }
  __syncthreads();

  // ---- row norms straight from LDS (threads 0..127 -> A, 128..255 -> B) ----
  {
    const int r = t & 127;
    const float* src = (t < 128) ? &sA[r * LSTRIDE] : &sB[r * LSTRIDE];
    float s = 0.f;
#pragma unroll 8
    for (int k = 0; k < DDIM; ++k) s += src[k] * src[k];
    if (t < 128) nA[r] = s; else nB[r] = s;
  }
  __syncthreads();

  const int wave  = t >> 5;
  const int lane  = t & 31;
  const int m0    = wave * 16;          // this wave's 16 A-rows
  const int lr    = lane & 15;
  const int khalf = (lane >> 4) * 2;    // K-split across lane halves (ISA layout)

  v8f acc[8];
  const v8f vzero = {};
#pragma unroll
  for (int j = 0; j < 8; ++j) acc[j] = vzero;

  const float* aRow = &sA[(m0 + lr) * LSTRIDE];
#pragma unroll 4
  for (int k0 = 0; k0 < DDIM; k0 += 4) {
    const int kb = k0 + khalf;
    const v2f af = *(const v2f*)(aRow + kb);   // A 16x4 fragment
#pragma unroll
    for (int j = 0; j < 8; ++j) {
      const v2f bf = *(const v2f*)(&sB[(j * 16 + lr) * LSTRIDE] + kb);  // B 4x16
      acc[j] = __builtin_amdgcn_wmma_f32_16x16x4_f32(
          false, af, false, bf, (short)0, acc[j], false, false);
    }
  }

  // ---- epilogue: sq = |a|^2 + |b|^2 - 2*dot ; relu(1 - sqrt(max(sq,0))) ----
  const int mhi = (lane >> 4) * 8;  // C/D layout: VGPR r -> M = r + 8*(lane>=16)
  float na[8];
#pragma unroll
  for (int r = 0; r < 8; ++r) na[r] = nA[m0 + mhi + r];

  float sum = 0.f;
#pragma unroll
  for (int j = 0; j < 8; ++j) {
    const float nb = nB[j * 16 + lr];
#pragma unroll
    for (int r = 0; r < 8; ++r) {
      const float sq = na[r] + nb - 2.0f * acc[j][r];
      const float d  = sqrtf(fmaxf(sq, 0.0f));
      sum += fmaxf(1.0f - d, 0.0f);
    }
  }

  // ---- deterministic block reduction ----
  rbuf[t] = sum;
  __syncthreads();
  for (int w = 128; w > 0; w >>= 1) {
    if (t < w) rbuf[t] += rbuf[t + w];
    __syncthreads();
  }
  if (t == 0) {
    const int bid = (p * gridDim.y + blockIdx.y) * gridDim.x + blockIdx.x;
    partials[bid] = rbuf[0];
  }
}

// ---------------------------------------------------------------------------
// Kernel 5: final deterministic reduction + mean.
// ---------------------------------------------------------------------------
__global__ __launch_bounds__(256) void k_final(const float* __restrict__ partials,
                                               float* __restrict__ out, int n) {
  __shared__ float buf[256];
  const int t = threadIdx.x;
  float s = 0.f;
  for (int i = t; i < n; i += 256) s += partials[i];
  buf[t] = s;
  __syncthreads();
  for (int w = 128; w > 0; w >>= 1) {
    if (t < w) buf[t] += buf[t + w];
    __syncthreads();
  }
  if (t == 0) out[0] = buf[0] * (1.0f / (31.0f * 1024.0f * 1024.0f));
}

// ---------------------------------------------------------------------------
extern "C" void kernel_launch(void* const* d_in, const int* in_sizes, int n_in,
                              void* d_out, int out_size, void* d_ws, size_t ws_size,
                              hipStream_t stream) {
  const float* reps  = (const float*)d_in[0];   // (32768, 128) fp32
  const int* labels  = (const int*)d_in[1];     // (32768,) int32
  float* out         = (float*)d_out;

  // workspace layout (ints): perm[32768] | cc[8192] | cb[8192] | cls[64 pad] | partials
  int* wsI      = (int*)d_ws;
  int* perm     = wsI;
  int* cc       = wsI + 32768;
  int* cb       = wsI + 32768 + NCHUNK * NCLS;
  int* cls      = wsI + 32768 + 2 * NCHUNK * NCLS;
  float* partials = (float*)(wsI + 32768 + 2 * NCHUNK * NCLS + 64);

  k_hist<<<1, 256, 0, stream>>>(labels, cc);
  k_scan<<<1, 32, 0, stream>>>(cc, cb, cls);
  k_scatter<<<1, 256, 0, stream>>>(labels, cb, perm);

  dim3 grid(MBLK / TILE, MBLK / TILE, NPAIR);   // (8, 8, 31)
  k_pairloss<<<grid, 256, 0, stream>>>(reps, perm, partials);

  k_final<<<1, 256, 0, stream>>>(partials, out, NPAIR * (MBLK / TILE) * (MBLK / TILE));
}